// CosineLayer_44367012168480
// MI455X (gfx1250) — compile-verified
//
#include <hip/hip_runtime.h>

typedef float v2f __attribute__((ext_vector_type(2)));
typedef float v8f __attribute__((ext_vector_type(8)));

#define K_DIM   768
#define M_ROWS  256
#define N_COLS  434056LL
#define NP1     434057LL
#define NFULL   434048LL            // 6782 full 64-column tiles
#define SUBSTRIDE (16 * K_DIM)      // floats between 16-row/col sub-tiles (49152 B)

// ---------------------------------------------------------------------------
// Kernel 1: L2-normalize the 256 feature rows into workspace (fhat), and
// write the constant threshold column out[:, N_COLS].
// ---------------------------------------------------------------------------
__global__ __launch_bounds__(256)
void normalize_features(const float* __restrict__ f,
                        const float* __restrict__ thr,
                        float* __restrict__ fhat,
                        float* __restrict__ out) {
    const int row = blockIdx.x;
    const int tid = threadIdx.x;

    const float* fr = f + (long long)row * K_DIM;
    float s = 0.0f;
    for (int k = tid; k < K_DIM; k += 256) {
        float v = fr[k];
        s = fmaf(v, v, s);
    }
    for (int off = 16; off > 0; off >>= 1)
        s += __shfl_xor(s, off, 32);

    __shared__ float red[8];
    __shared__ float invs;
    const int wave = tid >> 5, lane = tid & 31;
    if (lane == 0) red[wave] = s;
    __syncthreads();
    if (tid == 0) {
        float t = 0.0f;
        for (int i = 0; i < 8; ++i) t += red[i];
        invs = 1.0f / fmaxf(sqrtf(t), 1e-8f);
        out[(long long)row * NP1 + N_COLS] = thr[0];   // threshold column
    }
    __syncthreads();

    const float inv = invs;
    float* fo = fhat + (long long)row * K_DIM;
    for (int k = tid; k < K_DIM; k += 256)
        fo[k] = fr[k] * inv;
}

// ---------------------------------------------------------------------------
// Kernel 2 (hot): full 64-column tiles only.
//   out[m, n] = fhat[m,:] . w[n,:] / max(||w[n,:]||, eps)
// Block = 256 threads = 8 waves covering ALL 256 feature rows:
//   wave w -> rows [w*32, w*32+32), cols [blockIdx.x*64, +64)
// Per wave: 32M x 64N = 8 fp32-WMMA accumulators (2 A frags x 4 B sub-tiles).
// Each weight byte is fetched by exactly one block (8-wave reuse via WGP$).
// All A/B fragments are immediate-offset global_load_b64 off two pointers.
// Weight-row sum-of-squares fused as one v_pk_fma_f32 per B fragment.
// ---------------------------------------------------------------------------
__global__ __launch_bounds__(256)
void cosine_gemm_wmma(const float* __restrict__ fhat,
                      const float* __restrict__ w,
                      float* __restrict__ out) {
    const int lane = threadIdx.x & 31;
    const int wave = threadIdx.x >> 5;       // 0..7
    const int half = lane >> 4;              // 0 / 1
    const int l15  = lane & 15;
    const int kOff = half * 2;               // K offset this half-wave owns

    const int       mbase = wave * 32;                      // 0..224
    const long long nbase = (long long)blockIdx.x * 64;     // < NFULL

    // A: rows (mbase + l15) and (mbase + 16 + l15); second frag at +SUBSTRIDE
    const float* ap = fhat + (long long)(mbase + l15) * K_DIM + kOff;
    // B: column (nbase + l15) for sub-tile 0; sub-tile t at +t*SUBSTRIDE
    const float* bp = w + (nbase + l15) * (long long)K_DIM + kOff;

    v8f acc0[4] = {};        // rows mbase..mbase+15
    v8f acc1[4] = {};        // rows mbase+16..mbase+31
    v2f ssq[4]  = {};        // per-lane partial sum of squares (pair over K%4)

#pragma unroll 2
    for (int k = 0; k < K_DIM; k += 4) {
        v2f a0 = *(const v2f*)(ap + 0 * SUBSTRIDE);
        v2f a1 = *(const v2f*)(ap + 1 * SUBSTRIDE);
        ap += 4;
        v2f b0 = *(const v2f*)(bp + 0 * SUBSTRIDE);
        v2f b1 = *(const v2f*)(bp + 1 * SUBSTRIDE);
        v2f b2 = *(const v2f*)(bp + 2 * SUBSTRIDE);
        v2f b3 = *(const v2f*)(bp + 3 * SUBSTRIDE);
        bp += 4;

        acc0[0] = __builtin_amdgcn_wmma_f32_16x16x4_f32(false, a0, false, b0, (short)0, acc0[0], false, false);
        acc0[1] = __builtin_amdgcn_wmma_f32_16x16x4_f32(false, a0, false, b1, (short)0, acc0[1], false, false);
        acc0[2] = __builtin_amdgcn_wmma_f32_16x16x4_f32(false, a0, false, b2, (short)0, acc0[2], false, false);
        acc0[3] = __builtin_amdgcn_wmma_f32_16x16x4_f32(false, a0, false, b3, (short)0, acc0[3], false, false);
        acc1[0] = __builtin_amdgcn_wmma_f32_16x16x4_f32(false, a1, false, b0, (short)0, acc1[0], false, false);
        acc1[1] = __builtin_amdgcn_wmma_f32_16x16x4_f32(false, a1, false, b1, (short)0, acc1[1], false, false);
        acc1[2] = __builtin_amdgcn_wmma_f32_16x16x4_f32(false, a1, false, b2, (short)0, acc1[2], false, false);
        acc1[3] = __builtin_amdgcn_wmma_f32_16x16x4_f32(false, a1, false, b3, (short)0, acc1[3], false, false);

        ssq[0] += b0 * b0;   // v_pk_fma_f32
        ssq[1] += b1 * b1;
        ssq[2] += b2 * b2;
        ssq[3] += b3 * b3;
    }

#pragma unroll
    for (int t = 0; t < 4; ++t) {
        // this lane covered K = kOff, kOff+1 (mod 4); partner lane the rest
        float s   = ssq[t].x + ssq[t].y;
        s        += __shfl_xor(s, 16, 32);
        float inv = 1.0f / fmaxf(sqrtf(s), 1e-8f);

        const long long col = nbase + t * 16 + l15;
        float* o0 = out + (long long)(mbase + half * 8) * NP1 + col;
#pragma unroll
        for (int r = 0; r < 8; ++r)   // rows mbase + r + 8*half
            __builtin_nontemporal_store(acc0[t][r] * inv, o0 + (long long)r * NP1);
        float* o1 = o0 + 16 * NP1;
#pragma unroll
        for (int r = 0; r < 8; ++r)   // rows mbase + 16 + r + 8*half
            __builtin_nontemporal_store(acc1[t][r] * inv, o1 + (long long)r * NP1);
    }
}

// ---------------------------------------------------------------------------
// Kernel 3: ragged edge — last 8 columns (434048..434055), all 256 rows.
// One wave per (row, col) dot product; trivial traffic (~24 KB re-read).
// ---------------------------------------------------------------------------
__global__ __launch_bounds__(256)
void edge_columns(const float* __restrict__ fhat,
                  const float* __restrict__ w,
                  float* __restrict__ out) {
    const int wave = threadIdx.x >> 5, lane = threadIdx.x & 31;
    const int idx = blockIdx.x * 8 + wave;          // 0..2047
    const int m  = idx >> 3;
    const long long c = NFULL + (idx & 7);          // 434048..434055

    const float* fr = fhat + (long long)m * K_DIM;
    const float* wr = w + c * K_DIM;
    float dot = 0.0f, ss = 0.0f;
    for (int k = lane; k < K_DIM; k += 32) {
        float wv = wr[k];
        dot = fmaf(fr[k], wv, dot);
        ss  = fmaf(wv,  wv, ss);
    }
    for (int off = 16; off > 0; off >>= 1) {
        dot += __shfl_xor(dot, off, 32);
        ss  += __shfl_xor(ss,  off, 32);
    }
    if (lane == 0)
        out[(long long)m * NP1 + c] = dot / fmaxf(sqrtf(ss), 1e-8f);
}

// ---------------------------------------------------------------------------
extern "C" void kernel_launch(void* const* d_in, const int* in_sizes, int n_in,
                              void* d_out, int out_size, void* d_ws, size_t ws_size,
                              hipStream_t stream) {
    const float* features = (const float*)d_in[0];   // [256, 768] f32
    const float* weight   = (const float*)d_in[1];   // [434056, 768] f32
    const float* thr      = (const float*)d_in[2];   // scalar f32
    float*       out      = (float*)d_out;           // [256, 434057] f32
    float*       fhat     = (float*)d_ws;            // 768 KiB scratch

    normalize_features<<<dim3(M_ROWS), dim3(256), 0, stream>>>(features, thr, fhat, out);

    const unsigned ntiles = (unsigned)(NFULL / 64);  // 6782 full tiles
    cosine_gemm_wmma<<<dim3(ntiles, 1, 1), dim3(256), 0, stream>>>(fhat, weight, out);

    edge_columns<<<dim3(256), dim3(256), 0, stream>>>(fhat, weight, out);
}